// GCNTemporalPredictor_15874199126537
// MI455X (gfx1250) — compile-verified
//
#include <hip/hip_runtime.h>
#include <hip/hip_bf16.h>

typedef __attribute__((ext_vector_type(2))) float v2f;
typedef __attribute__((ext_vector_type(8))) float v8f;

// ---------------------------------------------------------------------------
// Small utility kernels
// ---------------------------------------------------------------------------
__global__ void zero_f32(float* __restrict__ p, int n) {
  int i = blockIdx.x * blockDim.x + threadIdx.x;
  if (i < n) p[i] = 0.0f;
}

// Build Wcat[128][128] = [[Wxz0;Wxz1] | [Wxh0;Wxh1]] and combined biases.
__global__ void build_wcat(const float* __restrict__ Wxz0, const float* __restrict__ Wxz1,
                           const float* __restrict__ Wxh0, const float* __restrict__ Wxh1,
                           const float* __restrict__ bxz, const float* __restrict__ bhz,
                           const float* __restrict__ bxh, const float* __restrict__ bhh,
                           float* __restrict__ Wcat, float* __restrict__ bz,
                           float* __restrict__ bh) {
  int i = blockIdx.x * blockDim.x + threadIdx.x;
  if (i < 128 * 128) {
    int k = i >> 7, j = i & 127;
    float v;
    if (j < 64) v = (k < 64) ? Wxz0[k * 64 + j] : Wxz1[(k - 64) * 64 + j];
    else        v = (k < 64) ? Wxh0[k * 64 + (j - 64)] : Wxh1[(k - 64) * 64 + (j - 64)];
    Wcat[i] = v;
  }
  if (i < 64) {
    bz[i] = bxz[i] + bhz[i];
    bh[i] = bxh[i] + bhh[i];
  }
}

// deg = segment_sum(edge_weight, src)
__global__ void deg_kernel(const long long* __restrict__ src, const float* __restrict__ w,
                           float* __restrict__ deg, int E) {
  int i = blockIdx.x * blockDim.x + threadIdx.x;
  if (i < E) atomicAdd(&deg[src[i]], w[i]);
}

__global__ void dinv_kernel(const float* __restrict__ deg, float* __restrict__ dinv, int n) {
  int i = blockIdx.x * blockDim.x + threadIdx.x;
  if (i < n) {
    float dg = deg[i];
    dinv[i] = (dg > 0.0f) ? rsqrtf(dg) : 0.0f;
  }
}

// Px[dst] += (-dinv[src]*w*dinv[dst]) * x[src]   (one wave32 per edge, 2 feats/lane)
__global__ void scatter_kernel(const long long* __restrict__ src, const long long* __restrict__ dst,
                               const float* __restrict__ w, const float* __restrict__ dinv,
                               const float* __restrict__ x, float* __restrict__ Px, int E) {
  int gid = blockIdx.x * blockDim.x + threadIdx.x;
  int eid = gid >> 5;
  int lane = gid & 31;
  if (eid >= E) return;
  long long s = src[eid];
  long long d = dst[eid];
  float coef = -dinv[s] * w[eid] * dinv[d];
  const float* xr = x + s * 64;
  float* pr = Px + d * 64;
  atomicAdd(&pr[lane],      coef * xr[lane]);
  atomicAdd(&pr[lane + 32], coef * xr[lane + 32]);
}

// ---------------------------------------------------------------------------
// Fused WMMA kernel: [x|Px] @ Wcat -> Z,Ht -> H=(1-Z)*Ht -> H@Wlin+blin -> L2norm
// 128 threads (4 waves), 16 rows per block.
// ---------------------------------------------------------------------------
__global__ __launch_bounds__(128) void gemm_fused(
    const float* __restrict__ x, const float* __restrict__ Px,
    const float* __restrict__ Wcat, const float* __restrict__ bz,
    const float* __restrict__ bh, const float* __restrict__ Wlin,
    const float* __restrict__ blin, float* __restrict__ out) {
  __shared__ float Atile[16][132];  // [x|Px] rows, padded
  __shared__ float Hz[16][66];      // sigmoid(z) tiles, padded
  __shared__ float Hh[16][68];      // tanh(h) then H=(1-Z)*Ht, padded
  __shared__ float O[16][33];       // 2nd-gemm output, padded
  __shared__ float rinv[16];

  const int t    = threadIdx.x;
  const int w    = t >> 5;        // wave id 0..3
  const int lane = t & 31;
  const int row0 = blockIdx.x << 4;

  // Stage A = [x | Px] 16x128 tile into LDS (coalesced halves)
  for (int i = 0; i < 16; ++i) {
    float v = (t < 64) ? x[(row0 + i) * 64 + t] : Px[(row0 + i) * 64 + (t - 64)];
    Atile[i][t] = v;
  }
  __syncthreads();

  const int ln16  = lane & 15;             // M for A-frag, N for B-frag, N for C/D
  const int koff  = (lane >> 4) << 1;      // 0 or 2
  const int mbase = (lane >> 4) << 3;      // C/D row base: 0 or 8
  const int nblk  = w << 5;                // this wave's 32 output columns

  v8f acc0 = {0.f, 0.f, 0.f, 0.f, 0.f, 0.f, 0.f, 0.f};
  v8f acc1 = {0.f, 0.f, 0.f, 0.f, 0.f, 0.f, 0.f, 0.f};

  for (int k0 = 0; k0 < 128; k0 += 4) {
    v2f a;
    a.x = Atile[ln16][k0 + koff];
    a.y = Atile[ln16][k0 + koff + 1];
    const float* wr0 = Wcat + (k0 + koff) * 128;
    const float* wr1 = Wcat + (k0 + koff + 1) * 128;
    v2f b0, b1;
    b0.x = wr0[nblk + ln16];       b0.y = wr1[nblk + ln16];
    b1.x = wr0[nblk + 16 + ln16];  b1.y = wr1[nblk + 16 + ln16];
    acc0 = __builtin_amdgcn_wmma_f32_16x16x4_f32(false, a, false, b0, (short)0, acc0, false, false);
    acc1 = __builtin_amdgcn_wmma_f32_16x16x4_f32(false, a, false, b1, (short)0, acc1, false, false);
  }

  // Epilogue 1: activations into LDS. Waves 0-1 => z-path cols [0,64),
  // waves 2-3 => h-path cols [64,128).
  {
    const int c0 = nblk + ln16;
    const int c1 = nblk + 16 + ln16;
    for (int q = 0; q < 8; ++q) {
      int r = mbase + q;
      float p0 = acc0[q];
      float p1 = acc1[q];
      if (w < 2) {
        Hz[r][c0] = 1.0f / (1.0f + expf(-(p0 + bz[c0])));
        Hz[r][c1] = 1.0f / (1.0f + expf(-(p1 + bz[c1])));
      } else {
        Hh[r][c0 - 64] = tanhf(p0 + bh[c0 - 64]);
        Hh[r][c1 - 64] = tanhf(p1 + bh[c1 - 64]);
      }
    }
  }
  __syncthreads();

  // H = (1 - Z) * Ht   (GRU with H0 = 0; R-gate is dead code)
  for (int i = t; i < 16 * 64; i += 128) {
    int r = i >> 6, c = i & 63;
    Hh[r][c] = (1.0f - Hz[r][c]) * Hh[r][c];
  }
  __syncthreads();

  // Second GEMM: H(16x64) @ Wlin(64x32) + blin. Waves 0,1 each own 16 columns.
  if (w < 2) {
    const int nb = w << 4;
    v8f acc = {0.f, 0.f, 0.f, 0.f, 0.f, 0.f, 0.f, 0.f};
    for (int k0 = 0; k0 < 64; k0 += 4) {
      v2f a;
      a.x = Hh[ln16][k0 + koff];
      a.y = Hh[ln16][k0 + koff + 1];
      v2f b;
      b.x = Wlin[(k0 + koff) * 32 + nb + ln16];
      b.y = Wlin[(k0 + koff + 1) * 32 + nb + ln16];
      acc = __builtin_amdgcn_wmma_f32_16x16x4_f32(false, a, false, b, (short)0, acc, false, false);
    }
    float bl = blin[nb + ln16];
    for (int q = 0; q < 8; ++q) O[mbase + q][nb + ln16] = acc[q] + bl;
  }
  __syncthreads();

  // Row L2 norm with eps clamp
  if (t < 16) {
    float s = 0.0f;
    for (int c = 0; c < 32; ++c) { float q = O[t][c]; s += q * q; }
    rinv[t] = 1.0f / fmaxf(sqrtf(s), 1e-12f);
  }
  __syncthreads();

  for (int i = t; i < 16 * 32; i += 128) {
    int r = i >> 5, c = i & 31;
    out[(row0 + r) * 32 + c] = O[r][c] * rinv[r];
  }
}

// ---------------------------------------------------------------------------
extern "C" void kernel_launch(void* const* d_in, const int* in_sizes, int n_in,
                              void* d_out, int out_size, void* d_ws, size_t ws_size,
                              hipStream_t stream) {
  const float*     x    = (const float*)d_in[0];
  const long long* eidx = (const long long*)d_in[1];   // int64 [2,E]
  const float*     ew   = (const float*)d_in[2];
  const float*     Wxz0 = (const float*)d_in[3];
  const float*     Wxz1 = (const float*)d_in[4];
  const float*     bxz  = (const float*)d_in[5];
  const float*     bhz  = (const float*)d_in[8];
  // indices 9..14 (r-gate) are provably unused: H0 == 0 -> H*R == 0
  const float*     Wxh0 = (const float*)d_in[15];
  const float*     Wxh1 = (const float*)d_in[16];
  const float*     bxh  = (const float*)d_in[17];
  const float*     bhh  = (const float*)d_in[20];
  const float*     Wlin = (const float*)d_in[21];
  const float*     blin = (const float*)d_in[22];

  const int N = in_sizes[0] / 64;
  const int E = in_sizes[2];
  const long long* src = eidx;
  const long long* dst = eidx + E;

  // Workspace carve-out (256B aligned slabs)
  char* ws = (char*)d_ws;
  size_t off = 0;
  auto carve = [&](size_t bytes) -> void* {
    void* p = ws + off;
    off = (off + bytes + 255) & ~(size_t)255;
    return p;
  };
  float* deg  = (float*)carve((size_t)N * sizeof(float));
  float* dinv = (float*)carve((size_t)N * sizeof(float));
  float* Px   = (float*)carve((size_t)N * 64 * sizeof(float));
  float* Wcat = (float*)carve(128 * 128 * sizeof(float));
  float* bz   = (float*)carve(64 * sizeof(float));
  float* bh   = (float*)carve(64 * sizeof(float));
  (void)ws_size; (void)n_in; (void)out_size;

  zero_f32<<<(N + 255) / 256, 256, 0, stream>>>(deg, N);
  zero_f32<<<((N * 64) + 255) / 256, 256, 0, stream>>>(Px, N * 64);
  build_wcat<<<(128 * 128 + 255) / 256, 256, 0, stream>>>(
      Wxz0, Wxz1, Wxh0, Wxh1, bxz, bhz, bxh, bhh, Wcat, bz, bh);
  deg_kernel<<<(E + 255) / 256, 256, 0, stream>>>(src, ew, deg, E);
  dinv_kernel<<<(N + 255) / 256, 256, 0, stream>>>(deg, dinv, N);
  scatter_kernel<<<(int)(((size_t)E * 32 + 255) / 256), 256, 0, stream>>>(
      src, dst, ew, dinv, x, Px, E);
  gemm_fused<<<N / 16, 128, 0, stream>>>(x, Px, Wcat, bz, bh, Wlin, blin, (float*)d_out);
}